// EnhancedGAT_61160334295347
// MI455X (gfx1250) — compile-verified
//
#include <hip/hip_runtime.h>
#include <math.h>

typedef float v2f __attribute__((ext_vector_type(2)));
typedef float v8f __attribute__((ext_vector_type(8)));
typedef unsigned int u4v __attribute__((ext_vector_type(4)));
typedef int i8v __attribute__((ext_vector_type(8)));
typedef int i4v __attribute__((ext_vector_type(4)));

constexpr int N_NODES = 50000;
constexpr int N_EDGES = 800000;
constexpr int N_GRAPH = 256;
constexpr int F_IN    = 128;
constexpr int HID     = 64;
constexpr int HEADS   = 4;
constexpr int HC      = HEADS * HID;          // 256
constexpr int ETOT    = N_EDGES + N_NODES;    // 850000 (with self loops)
constexpr float NEG_SLOPE = 0.2f;
constexpr float BN_EPS    = 1e-5f;

// ---------------------------------------------------------------- utilities
__global__ void fill_f32(float* __restrict__ p, float v, int n) {
  int i = blockIdx.x * blockDim.x + threadIdx.x;
  if (i < n) p[i] = v;
}

__device__ inline void atomicMaxF(float* addr, float v) {
  // monotonic-bit-pattern trick: works for any sign, init with -INF
  if (v >= 0.0f) atomicMax((int*)addr, __float_as_int(v));
  else           atomicMin((unsigned int*)addr, (unsigned int)__float_as_int(v));
}

__device__ inline void edge_sd(const int* __restrict__ ei, int e, int& s, int& d) {
  if (e < N_EDGES) { s = ei[e]; d = ei[N_EDGES + e]; }
  else             { s = e - N_EDGES; d = s; }          // self loop
}

// ---------------------------------------------------- self-loop attr (mean)
__global__ void loop_attr_accum(const int* __restrict__ ei, const float* __restrict__ ea,
                                float* __restrict__ cnt, float* __restrict__ lattr) {
  int e = blockIdx.x * blockDim.x + threadIdx.x;
  if (e >= N_EDGES) return;
  int d = ei[N_EDGES + e];
  atomicAdd(&cnt[d], 1.0f);
  atomicAdd(&lattr[d], ea[e]);
}

__global__ void loop_attr_div(float* __restrict__ lattr, const float* __restrict__ cnt) {
  int n = blockIdx.x * blockDim.x + threadIdx.x;
  if (n < N_NODES) lattr[n] /= fmaxf(cnt[n], 1.0f);
}

// ------------------------------------------------------- WMMA f32 GEMM + TDM
// C[M,Nc] = A[M,K] @ B[Nc,K]^T  (B row-major [Nc,K], i.e. the torch weight).
// One block per 16-row A panel. The panel (16 x K f32, K<=256 -> <=16KB LDS)
// is DMA'd into LDS once by the Tensor Data Mover; 4 waves then sweep the
// tn tiles, reading A from LDS (16x reuse for layer 1) and streaming B from
// global. K stepped by 4 with V_WMMA_F32_16X16X4_F32.
__global__ void gemm_wmma_f32(const float* __restrict__ A, const float* __restrict__ B,
                              float* __restrict__ C, int M, int K, int Nc) {
  __shared__ float sA[16 * 256];
  int lane = threadIdx.x & 31;
  int wave = threadIdx.x >> 5;
  int tm   = blockIdx.x;                      // row tile index, M/16 blocks

  if (wave == 0) {
    // ---- Tensor DMA descriptor (D#), 2-D tensor: rows of A, one 16 x K tile
    unsigned lds_off = (unsigned)(size_t)(void*)sA;   // low 32 bits = LDS offset
    unsigned long long ga =
        (unsigned long long)(const void*)(A + (size_t)tm * 16 * K);
    u4v g0;
    g0.x = 1u;                                        // count=1, user mode
    g0.y = lds_off;                                   // lds_addr (bytes)
    g0.z = (unsigned)(ga & 0xFFFFFFFFull);            // global_addr[31:0]
    g0.w = (unsigned)((ga >> 32) & 0x01FFFFFFull) | (2u << 30);  // [56:32]+type=2
    i8v g1;
    g1[0] = 0x20000;                                  // data_size=2 (4 bytes)
    g1[1] = (K & 0xFFFF) << 16;                       // tensor_dim0[15:0]
    g1[2] = ((K >> 16) & 0xFFFF) | ((M & 0xFFFF) << 16);  // dim0 hi | dim1 lo
    g1[3] = ((M >> 16) & 0xFFFF) | ((K & 0xFFFF) << 16);  // dim1 hi | tile_dim0=K
    g1[4] = 16;                                       // tile_dim1=16, tile_dim2=0
    g1[5] = K;                                        // tensor_dim0_stride[31:0]
    g1[6] = 0;                                        // stride hi, dim1_stride lo
    g1[7] = 0;
    i4v z4 = {0, 0, 0, 0};                            // groups 2/3 unused (2-D)
    i8v z8 = {0, 0, 0, 0, 0, 0, 0, 0};
    __builtin_amdgcn_tensor_load_to_lds(g0, g1, z4, z4, z8, 0);
    __builtin_amdgcn_s_wait_tensorcnt(0);
    __asm__ volatile("" ::: "memory");
  }
  __syncthreads();
  __asm__ volatile("" ::: "memory");

  int half = lane >> 4;   // lanes 16-31 hold the upper K pair / M+8 rows
  int l16  = lane & 15;
  int tilesN = Nc >> 4;
  // A 16x4 layout: lane&15 = M row, vgpr = K within pair, lane>>4 = K pair
  const float* aLds = sA + (size_t)l16 * K + half * 2;

  for (int tn = wave; tn < tilesN; tn += 4) {
    // B 4x16: lane&15 = N col, mirrored K layout; B[n][k] = Bmat[n*K+k]
    const float* brow = B + (size_t)(tn * 16 + l16) * K + half * 2;
    v8f acc = {};
    for (int k = 0; k < K; k += 4) {
      v2f a = *(const v2f*)(aLds + k);               // ds_load_b64
      v2f b = *(const v2f*)(brow + k);               // global_load_b64
      acc = __builtin_amdgcn_wmma_f32_16x16x4_f32(false, a, false, b,
                                                  (short)0, acc, false, false);
    }
    // D 16x16: vgpr i -> row tm*16 + i + 8*half, col tn*16 + (lane&15)
    int n_c = tn * 16 + l16;
    int m0  = tm * 16 + half * 8;
#pragma unroll
    for (int i = 0; i < 8; ++i)
      C[(size_t)(m0 + i) * Nc + n_c] = acc[i];
  }
}

// -------------------------------- per-head scalar: dot(We_head, att_e_head)
__global__ void edge_scalars(const float* __restrict__ We1, const float* __restrict__ ae1,
                             const float* __restrict__ We2, const float* __restrict__ ae2,
                             float* __restrict__ sedge) {
  int t = threadIdx.x;
  if (t < HEADS) {
    float s = 0.f;
    for (int c = 0; c < HID; ++c) s += We1[t * HID + c] * ae1[t * HID + c];
    sedge[t] = s;
  } else if (t == HEADS) {
    float s = 0.f;
    for (int c = 0; c < HID; ++c) s += We2[c] * ae2[c];
    sedge[HEADS] = s;
  }
}

// ----------------------------------------- a_src / a_dst node attention dots
__global__ void att_dots(const float* __restrict__ h, const float* __restrict__ att_s,
                         const float* __restrict__ att_d, float* __restrict__ asrc,
                         float* __restrict__ adst, int Hh, int C) {
  int i = blockIdx.x * blockDim.x + threadIdx.x;  // n*Hh + hd
  if (i >= N_NODES * Hh) return;
  int n = i / Hh, hd = i % Hh;
  const float* hp = h + (size_t)n * Hh * C + (size_t)hd * C;
  float s = 0.f, d = 0.f;
  for (int c = 0; c < C; ++c) {
    float v = hp[c];
    s += v * att_s[hd * C + c];
    d += v * att_d[hd * C + c];
  }
  asrc[i] = s;
  adst[i] = d;
}

// ------------------------------------------ alpha = leaky(...); segment max
__global__ void alpha_raw_max(const int* __restrict__ ei, const float* __restrict__ ea,
                              const float* __restrict__ lattr, const float* __restrict__ asrc,
                              const float* __restrict__ adst, const float* __restrict__ sedge,
                              float* __restrict__ ex, float* __restrict__ m, int Hh) {
  int e = blockIdx.x * blockDim.x + threadIdx.x;
  if (e >= ETOT) return;
  int s, d; edge_sd(ei, e, s, d);
  float eav = (e < N_EDGES) ? ea[e] : lattr[e - N_EDGES];
  for (int hd = 0; hd < Hh; ++hd) {
    float raw = asrc[s * Hh + hd] + adst[d * Hh + hd] + eav * sedge[hd];
    raw = raw > 0.f ? raw : NEG_SLOPE * raw;
    ex[(size_t)e * Hh + hd] = raw;
    atomicMaxF(&m[d * Hh + hd], raw);
  }
}

// ------------------------------------------------ exp(alpha-m); segment sum
__global__ void alpha_exp_sum(const int* __restrict__ ei, float* __restrict__ ex,
                              const float* __restrict__ m, float* __restrict__ ssum, int Hh) {
  int e = blockIdx.x * blockDim.x + threadIdx.x;
  if (e >= ETOT) return;
  int s, d; edge_sd(ei, e, s, d);
  for (int hd = 0; hd < Hh; ++hd) {
    float v = __expf(ex[(size_t)e * Hh + hd] - m[d * Hh + hd]);
    ex[(size_t)e * Hh + hd] = v;
    atomicAdd(&ssum[d * Hh + hd], v);
  }
}

// --------------------------- agg[dst, :] += (ex/s) * h[src, :]  (coalesced)
__global__ void aggregate(const int* __restrict__ ei, const float* __restrict__ ex,
                          const float* __restrict__ ssum, const float* __restrict__ h,
                          float* __restrict__ agg, int Hh, int C) {
  int e = blockIdx.x;             // one block per edge
  int c = threadIdx.x;            // Hh*C threads, contiguous channels
  int s, d; edge_sd(ei, e, s, d);
  int hd = c / C;
  float coeff = ex[(size_t)e * Hh + hd] / ssum[d * Hh + hd];
  atomicAdd(&agg[(size_t)d * Hh * C + c], coeff * h[(size_t)s * Hh * C + c]);
}

// ---------------------------------------------------- BatchNorm statistics
__global__ void bn_stats(const float* __restrict__ x, float* __restrict__ mean,
                         float* __restrict__ var, int C) {
  int j = blockIdx.x;             // one block per column
  __shared__ float sh[256], sh2[256];
  float s = 0.f, s2 = 0.f;
  for (int r = threadIdx.x; r < N_NODES; r += blockDim.x) {
    float v = x[(size_t)r * C + j];
    s += v; s2 += v * v;
  }
  sh[threadIdx.x] = s; sh2[threadIdx.x] = s2;
  __syncthreads();
  for (int st = blockDim.x >> 1; st > 0; st >>= 1) {
    if ((int)threadIdx.x < st) {
      sh[threadIdx.x] += sh[threadIdx.x + st];
      sh2[threadIdx.x] += sh2[threadIdx.x + st];
    }
    __syncthreads();
  }
  if (threadIdx.x == 0) {
    float mu = sh[0] / (float)N_NODES;
    mean[j] = mu;
    var[j] = sh2[0] / (float)N_NODES - mu * mu;
  }
}

__global__ void bn_norm_relu(const float* __restrict__ x, const float* __restrict__ mean,
                             const float* __restrict__ var, const float* __restrict__ g,
                             const float* __restrict__ be, float* __restrict__ y, int C) {
  int i = blockIdx.x * blockDim.x + threadIdx.x;
  if (i >= N_NODES * C) return;
  int j = i % C;
  float v = (x[i] - mean[j]) * rsqrtf(var[j] + BN_EPS) * g[j] + be[j];
  y[i] = v > 0.f ? v : 0.f;
}

// ------------------------------------------------------- global mean pool
__global__ void pool(const int* __restrict__ batch, const float* __restrict__ x,
                     float* __restrict__ pooled, float* __restrict__ gcnt) {
  int n = blockIdx.x;             // one block per node, HID threads
  int c = threadIdx.x;
  int b = batch[n];
  atomicAdd(&pooled[(size_t)b * HID + c], x[(size_t)n * HID + c]);
  if (c == 0) atomicAdd(&gcnt[b], 1.0f);
}

__global__ void fc(const float* __restrict__ pooled, const float* __restrict__ gcnt,
                   const float* __restrict__ Wfc, const float* __restrict__ bfc,
                   float* __restrict__ out) {
  int g = blockIdx.x * blockDim.x + threadIdx.x;
  if (g >= N_GRAPH) return;
  float inv = 1.0f / fmaxf(gcnt[g], 1.0f);
  float s = 0.f;
  for (int c = 0; c < HID; ++c) s += pooled[(size_t)g * HID + c] * Wfc[c];
  out[g] = s * inv + bfc[0];
}

// ===========================================================================
extern "C" void kernel_launch(void* const* d_in, const int* in_sizes, int n_in,
                              void* d_out, int out_size, void* d_ws, size_t ws_size,
                              hipStream_t stream) {
  (void)in_sizes; (void)n_in; (void)out_size; (void)ws_size;

  const float* x      = (const float*)d_in[0];
  const int*   ei     = (const int*)  d_in[1];   // [2,E] flat: src then dst
  const float* ea     = (const float*)d_in[2];
  const int*   batch  = (const int*)  d_in[3];
  const float* W1     = (const float*)d_in[4];   // [HC, F_IN]
  const float* We1    = (const float*)d_in[5];   // [HC, 1]
  const float* att_s1 = (const float*)d_in[6];   // [H, HID]
  const float* att_d1 = (const float*)d_in[7];
  const float* att_e1 = (const float*)d_in[8];
  // b1 (d_in[9]) cancels through BatchNorm -> skipped
  const float* g1     = (const float*)d_in[10];
  const float* be1    = (const float*)d_in[11];
  const float* W2     = (const float*)d_in[12];  // [HID, HC]
  const float* We2    = (const float*)d_in[13];  // [HID, 1]
  const float* att_s2 = (const float*)d_in[14];
  const float* att_d2 = (const float*)d_in[15];
  const float* att_e2 = (const float*)d_in[16];
  // b2 (d_in[17]) cancels through BatchNorm -> skipped
  const float* g2     = (const float*)d_in[18];
  const float* be2    = (const float*)d_in[19];
  const float* Wfc    = (const float*)d_in[20];  // [1, HID]
  const float* bfc    = (const float*)d_in[21];
  float* out = (float*)d_out;

  // ------------------------------------------------ workspace layout (f32)
  float* w = (float*)d_ws;
  size_t o = 0;
  float* cnt   = w + o; o += N_NODES;
  float* lattr = w + o; o += N_NODES;
  float* h1    = w + o; o += (size_t)N_NODES * HC;   // layer1 features / x2
  float* asrc1 = w + o; o += (size_t)N_NODES * HEADS;
  float* adst1 = w + o; o += (size_t)N_NODES * HEADS;
  float* m1    = w + o; o += (size_t)N_NODES * HEADS;
  float* s1    = w + o; o += (size_t)N_NODES * HEADS;
  float* ex1   = w + o; o += (size_t)ETOT * HEADS;
  float* agg1  = w + o; o += (size_t)N_NODES * HC;
  float* mean1 = w + o; o += HC;
  float* var1  = w + o; o += HC;
  float* h2    = w + o; o += (size_t)N_NODES * HID;  // layer2 features / x3
  float* asrc2 = w + o; o += N_NODES;
  float* adst2 = w + o; o += N_NODES;
  float* m2    = w + o; o += N_NODES;
  float* s2    = w + o; o += N_NODES;
  float* ex2   = w + o; o += ETOT;
  float* agg2  = w + o; o += (size_t)N_NODES * HID;
  float* mean2 = w + o; o += HID;
  float* var2  = w + o; o += HID;
  float* pooled = w + o; o += (size_t)N_GRAPH * HID;
  float* gcnt  = w + o; o += N_GRAPH;
  float* sedge = w + o; o += 8;

  auto blk = [](long long n, int b) { return (int)((n + b - 1) / b); };

  // ---- self-loop edge attr (mean of incoming)
  fill_f32<<<blk(N_NODES, 256), 256, 0, stream>>>(cnt, 0.f, N_NODES);
  fill_f32<<<blk(N_NODES, 256), 256, 0, stream>>>(lattr, 0.f, N_NODES);
  loop_attr_accum<<<blk(N_EDGES, 256), 256, 0, stream>>>(ei, ea, cnt, lattr);
  loop_attr_div<<<blk(N_NODES, 256), 256, 0, stream>>>(lattr, cnt);

  // ---- layer 1: h1 = x @ W1^T  (TDM-staged A panel + WMMA)
  gemm_wmma_f32<<<N_NODES / 16, 128, 0, stream>>>(x, W1, h1, N_NODES, F_IN, HC);

  edge_scalars<<<1, 8, 0, stream>>>(We1, att_e1, We2, att_e2, sedge);
  att_dots<<<blk((long long)N_NODES * HEADS, 256), 256, 0, stream>>>(
      h1, att_s1, att_d1, asrc1, adst1, HEADS, HID);

  fill_f32<<<blk((long long)N_NODES * HEADS, 256), 256, 0, stream>>>(m1, -INFINITY, N_NODES * HEADS);
  fill_f32<<<blk((long long)N_NODES * HEADS, 256), 256, 0, stream>>>(s1, 0.f, N_NODES * HEADS);
  alpha_raw_max<<<blk(ETOT, 256), 256, 0, stream>>>(ei, ea, lattr, asrc1, adst1, sedge, ex1, m1, HEADS);
  alpha_exp_sum<<<blk(ETOT, 256), 256, 0, stream>>>(ei, ex1, m1, s1, HEADS);

  fill_f32<<<blk((long long)N_NODES * HC, 256), 256, 0, stream>>>(agg1, 0.f, N_NODES * HC);
  aggregate<<<ETOT, HC, 0, stream>>>(ei, ex1, s1, h1, agg1, HEADS, HID);

  bn_stats<<<HC, 256, 0, stream>>>(agg1, mean1, var1, HC);
  bn_norm_relu<<<blk((long long)N_NODES * HC, 256), 256, 0, stream>>>(
      agg1, mean1, var1, g1, be1, h1 /*becomes x2*/, HC);

  // ---- layer 2: h2 = x2 @ W2^T  (TDM-staged A panel + WMMA)
  gemm_wmma_f32<<<N_NODES / 16, 128, 0, stream>>>(h1, W2, h2, N_NODES, HC, HID);

  att_dots<<<blk(N_NODES, 256), 256, 0, stream>>>(h2, att_s2, att_d2, asrc2, adst2, 1, HID);

  fill_f32<<<blk(N_NODES, 256), 256, 0, stream>>>(m2, -INFINITY, N_NODES);
  fill_f32<<<blk(N_NODES, 256), 256, 0, stream>>>(s2, 0.f, N_NODES);
  alpha_raw_max<<<blk(ETOT, 256), 256, 0, stream>>>(ei, ea, lattr, asrc2, adst2, sedge + HEADS, ex2, m2, 1);
  alpha_exp_sum<<<blk(ETOT, 256), 256, 0, stream>>>(ei, ex2, m2, s2, 1);

  fill_f32<<<blk((long long)N_NODES * HID, 256), 256, 0, stream>>>(agg2, 0.f, N_NODES * HID);
  aggregate<<<ETOT, HID, 0, stream>>>(ei, ex2, s2, h2, agg2, 1, HID);

  bn_stats<<<HID, 256, 0, stream>>>(agg2, mean2, var2, HID);
  bn_norm_relu<<<blk((long long)N_NODES * HID, 256), 256, 0, stream>>>(
      agg2, mean2, var2, g2, be2, h2 /*becomes x3*/, HID);

  // ---- global mean pool + fc
  fill_f32<<<blk((long long)N_GRAPH * HID, 256), 256, 0, stream>>>(pooled, 0.f, N_GRAPH * HID);
  fill_f32<<<1, 256, 0, stream>>>(gcnt, 0.f, N_GRAPH);
  pool<<<N_NODES, HID, 0, stream>>>(batch, h2, pooled, gcnt);
  fc<<<1, 256, 0, stream>>>(pooled, gcnt, Wfc, bfc, out);
}